// HolonomyLayer_55602646614321
// MI455X (gfx1250) — compile-verified
//
#include <hip/hip_runtime.h>

// ---------------------------------------------------------------------------
// HolonomyLayer on MI455X (gfx1250): Wilson-loop traces via fp32 WMMA chains.
//   B=4, S=2048, D=64, features F=4 (loop sizes {2,4} x 2 loops).
//
// One workgroup (8 wave32) per 8 consecutive positions: the 14-matrix G halo
// [p0-2 .. p0+11] is DMA'd into LDS once with GLOBAL_LOAD_ASYNC_TO_LDS_B128
// (ASYNCcnt), amortizing halo reads 4x vs one-position blocks (kernel is
// L2-bandwidth bound; connection = 134 MB fits the 192 MB L2).
//
// Chain step: H <- H + c * (G @ H), a 64x64x64 fp32 matmul done with
// V_WMMA_F32_16X16X4_F32 (4 M-tiles x 2 N-tiles x 16 K-slices = 128 WMMA).
// Each wave owns feature f = wave>>1 and a private 32-column slab of H
// (column j of H_new depends only on column j of H), so the chain needs no
// inter-wave barriers; H lives in VGPRs (C layout) + a private row-major LDS
// copy for B-fragment reloads.
// ---------------------------------------------------------------------------

typedef __attribute__((ext_vector_type(2))) float v2f;
typedef __attribute__((ext_vector_type(8))) float v8f;

#define S_LEN  2048
#define B_N    4
#define D      64
#define POS    8                      // positions per workgroup
#define NGH    (POS + 6)              // halo matrices: p0-2 .. p0+POS+3
#define GELEMS (D * D)                // 4096 floats per matrix
#define GBYTES (GELEMS * 4)           // 16 KB per matrix
#define HCOLS  32                     // column slab per wave

// ---- one chain step on this wave's 64x32 slab: H += c * (G @ H) -----------
__device__ __forceinline__ void chain_step(const float* __restrict__ G,
                                           float* __restrict__ Hlds,
                                           v8f (&Hreg)[4][2],
                                           float c, int lo, int hi)
{
    v8f acc[4][2];
#pragma unroll
    for (int mt = 0; mt < 4; ++mt)
#pragma unroll
        for (int nt = 0; nt < 2; ++nt)
#pragma unroll
            for (int r = 0; r < 8; ++r) acc[mt][nt][r] = 0.0f;

#pragma unroll
    for (int kt = 0; kt < 16; ++kt) {
        // A fragments of G: 16x4 tile (mt, kt); contiguous K-pair per lane
        // in row-major G -> ds_load_b64.
        v2f a[4];
#pragma unroll
        for (int mt = 0; mt < 4; ++mt) {
            const int row = 16 * mt + lo;
            const int col = 4 * kt + 2 * hi;
            a[mt] = *(const v2f*)(G + row * D + col);
        }
        // B fragments of H slab: 4x16 tile (kt, nt) from row-major LDS slab.
        v2f bf[2];
#pragma unroll
        for (int nt = 0; nt < 2; ++nt) {
            const int r0 = 4 * kt + 2 * hi;
            const int cb = nt * 16 + lo;
            v2f t;
            t[0] = Hlds[(r0 + 0) * HCOLS + cb];
            t[1] = Hlds[(r0 + 1) * HCOLS + cb];
            bf[nt] = t;
        }
#pragma unroll
        for (int mt = 0; mt < 4; ++mt)
#pragma unroll
            for (int nt = 0; nt < 2; ++nt)
                acc[mt][nt] = __builtin_amdgcn_wmma_f32_16x16x4_f32(
                    false, a[mt], false, bf[nt], (short)0, acc[mt][nt],
                    false, false);
    }

    // H_new = H + c*acc ; update register copy and LDS slab (same-wave LDS
    // ops are in-order, so these stores cannot pass the fragment loads).
#pragma unroll
    for (int mt = 0; mt < 4; ++mt)
#pragma unroll
        for (int nt = 0; nt < 2; ++nt)
#pragma unroll
            for (int r = 0; r < 8; ++r) {
                const float v = Hreg[mt][nt][r] + c * acc[mt][nt][r];
                Hreg[mt][nt][r] = v;
                const int row  = 16 * mt + r + 8 * hi;
                const int lcol = nt * 16 + lo;
                Hlds[row * HCOLS + lcol] = v;
            }
}

extern "C" __global__ __launch_bounds__(256)
void holonomy_wmma_kernel(const float* __restrict__ conn,   // (B,S,64,64)
                          const float* __restrict__ proj,   // (4,64)
                          const float* __restrict__ temp,   // (1,)
                          float* __restrict__ out)          // (B,S,64)
{
    extern __shared__ float smem[];
    float* Gs    = smem;                        // NGH * 4096 floats (224 KB)
    float* Hs    = Gs + NGH * GELEMS;           // 8 waves * 64*32   (64 KB)
    float* sFeat = Hs + 8 * D * HCOLS;          // POS*4 feature accumulators

    const int tid    = threadIdx.x;
    const int blk    = blockIdx.x;              // b*(S/POS) + p0/POS
    const int b      = blk / (S_LEN / POS);
    const int p0     = (blk - b * (S_LEN / POS)) * POS;
    const int tbase  = p0 - 2;

    if (tid < POS * 4) sFeat[tid] = 0.0f;

    // ---- async DMA of the G halo straight into LDS (ASYNCcnt path) --------
    {
        const unsigned ldsG = (unsigned)(size_t)(void*)Gs; // LDS byte address
#pragma unroll 1
        for (int t = 0; t < NGH; ++t) {
            int j = tbase + t;
            j = j < 0 ? 0 : (j > S_LEN - 1 ? S_LEN - 1 : j);
            const char* src =
                (const char*)(conn + (size_t)(b * S_LEN + j) * GELEMS);
            const unsigned dbase = ldsG + (unsigned)t * (unsigned)GBYTES;
#pragma unroll 1
            for (int e = tid; e < GBYTES / 16; e += 256) {
                const unsigned voff = (unsigned)e * 16u;
                const unsigned dst  = dbase + voff;
                asm volatile(
                    "global_load_async_to_lds_b128 %0, %1, %2"
                    :
                    : "v"(dst), "v"(voff), "s"(src)
                    : "memory");
            }
        }
        asm volatile("s_wait_asynccnt 0" ::: "memory");
    }
    __syncthreads();

    // ---- per-wave task: feature f, column half ch (uniform / scalar) ------
    const int wave = __builtin_amdgcn_readfirstlane(tid >> 5);
    const int lane = tid & 31;
    const int lo = lane & 15;
    const int hi = lane >> 4;

    const int f  = wave >> 1;                   // 0..3
    const int ch = wave & 1;                    // columns ch*32 .. ch*32+31

    const int half   = (f < 2) ? 1 : 2;
    const int offset = (f == 1) ? 1 : ((f == 3) ? 2 : 0);

    float* Hlds = Hs + wave * (D * HCOLS);

#pragma clang loop unroll(disable)
    for (int q = 0; q < POS; ++q) {
        const int p   = p0 + q;
        const int adj = p + offset;
        int start = adj - half;     if (start < 0)       start = 0;
        int end   = adj + half;     if (end > S_LEN - 1) end   = S_LEN - 1;

        v8f Hreg[4][2];

        // init: H = I - 0.1*G[start] (first forward step folded in)
        {
            const float m0 = (start < end) ? 0.1f : 0.0f;
            const float* G0 = Gs + (start - tbase) * GELEMS;
#pragma unroll
            for (int mt = 0; mt < 4; ++mt)
#pragma unroll
                for (int nt = 0; nt < 2; ++nt)
#pragma unroll
                    for (int r = 0; r < 8; ++r) {
                        const int row  = 16 * mt + r + 8 * hi;
                        const int lcol = nt * 16 + lo;
                        const int gcol = ch * HCOLS + lcol;
                        const float v = (row == gcol ? 1.0f : 0.0f)
                                      - m0 * G0[row * D + gcol];
                        Hreg[mt][nt][r] = v;
                        Hlds[row * HCOLS + lcol] = v;
                    }
        }

        // forward leg: step = I - 0.1*G[i], i = start+k, active iff i < end
        for (int k = 1; k < 2 * half; ++k) {
            const int i = start + k;
            if (i < end)
                chain_step(Gs + (i - tbase) * GELEMS, Hlds, Hreg,
                           -0.1f, lo, hi);
        }
        // backward leg: step = I + 0.1*G[i], i = end-k, active iff i > start
        for (int k = 0; k < 2 * half; ++k) {
            const int i = end - k;
            if (i > start)
                chain_step(Gs + (i - tbase) * GELEMS, Hlds, Hreg,
                           +0.1f, lo, hi);
        }

        // trace of this wave's slab -> per-(position, feature) accumulator
        float part = 0.0f;
#pragma unroll
        for (int mt = 0; mt < 4; ++mt)
#pragma unroll
            for (int nt = 0; nt < 2; ++nt)
#pragma unroll
                for (int r = 0; r < 8; ++r) {
                    const int row  = 16 * mt + r + 8 * hi;
                    const int gcol = ch * HCOLS + nt * 16 + lo;
                    if (row == gcol) part += Hreg[mt][nt][r];
                }
        atomicAdd(&sFeat[q * 4 + f], part);     // ds_add_f32
    }
    __syncthreads();

    // ---- project 4 features -> 64 outputs for each of the POS positions ---
    const float invT = 1.0f / temp[0];
#pragma unroll
    for (int o = tid; o < POS * D; o += 256) {
        const int q = o >> 6;
        const int d = o & (D - 1);
        float v = 0.0f;
#pragma unroll
        for (int ff = 0; ff < 4; ++ff) v += sFeat[q * 4 + ff] * proj[ff * D + d];
        out[(size_t)(b * S_LEN + p0 + q) * D + d] = v * invT;
    }
}

extern "C" void kernel_launch(void* const* d_in, const int* in_sizes, int n_in,
                              void* d_out, int out_size, void* d_ws, size_t ws_size,
                              hipStream_t stream) {
    (void)in_sizes; (void)n_in; (void)out_size; (void)d_ws; (void)ws_size;
    // inputs: [0]=embeddings (unused by reference), [1]=connection,
    //         [2]=output_projection, [3]=integration_temperature
    const float* conn = (const float*)d_in[1];
    const float* proj = (const float*)d_in[2];
    const float* temp = (const float*)d_in[3];
    float*       out  = (float*)d_out;

    const size_t lds_bytes =
        (size_t)(NGH * GELEMS + 8 * D * HCOLS + POS * 4 + 8) * sizeof(float);
    // ~295 KB of the 320 KB WGP LDS

    holonomy_wmma_kernel<<<B_N * (S_LEN / POS), 256, lds_bytes, stream>>>(
        conn, proj, temp, out);
}